// InferPositionsPyroModel_13151189860535
// MI455X (gfx1250) — compile-verified
//
#include <hip/hip_runtime.h>
#include <math.h>

typedef __attribute__((ext_vector_type(2))) float v2f;
typedef __attribute__((ext_vector_type(8))) float v8f;

#define INV_2PI 0.15915494309189535f   // 1/(2*pi)
#define LOG2E   1.4426950408889634f

// out[m,n] = eps_perm[m]*eps[m]*rho/(2*pi*sigma) * exp(-||nuc_m - sb_n||^2/(2*sigma))
//          + eps[m]*d_drop[m]*chi[n]
//
// sq_dist 16x16 tile = one V_WMMA_F32_16X16X4_F32 (rank-4 trick):
//   A[m][k] = [-2x_m, -2y_m, |n_m|^2, 1]   (16x4)
//   B[k][n] = [ x_n ,  y_n , 1, |s_n|^2]   (4x16)
// Each wave owns one 16-row tile and sweeps CTW column tiles, keeping the A
// fragment and all per-row coefficients resident in VGPRs.
template <int MC>
__global__ void __launch_bounds__(256)
poisson_rate_kernel(const float* __restrict__ nuc,
                    const float* __restrict__ sb,
                    const float* __restrict__ eps_perm,
                    const float* __restrict__ eps,
                    const float* __restrict__ d_drop,
                    const float* __restrict__ rho,
                    const float* __restrict__ sigma,
                    const float* __restrict__ chi,
                    float* __restrict__ out,
                    int M_rt, int colTiles, int colGroups, int totalWaves)
{
    constexpr int CTW = 8;                         // column tiles per wave
    const int M = (MC > 0) ? MC : M_rt;            // compile-time stride when known

    const int lane = threadIdx.x & 31;
    const int wave = threadIdx.x >> 5;
    const int wg   = blockIdx.x * (blockDim.x >> 5) + wave;
    if (wg >= totalWaves) return;                  // wave-uniform

    const int rowTile  = wg / colGroups;
    const int colGroup = wg - rowTile * colGroups;
    const int laneHi   = lane >> 4;                // 0: lanes 0-15, 1: lanes 16-31
    const int laneLo   = lane & 15;

    // ---- A fragment (16x4 f32), built once per wave.
    //      lanes 0-15 hold K=0,1 ; lanes 16-31 hold K=2,3 (branchless cndmask)
    const int   mA = rowTile * 16 + laneLo;
    const float ax = nuc[2 * mA + 0];
    const float ay = nuc[2 * mA + 1];
    v2f a;
    a.x = laneHi ? (ax * ax + ay * ay) : (-2.0f * ax);   // K=2 : K=0
    a.y = laneHi ? 1.0f                : (-2.0f * ay);   // K=3 : K=1

    // ---- per-wave scalars: one v_rcp_f32 instead of two IEEE divides
    const float rs    = __builtin_amdgcn_rcpf(sigma[0]);
    const float scale = -0.5f * rs * LOG2E;              // sq_dist -> exp2 argument
    const float norm  = rho[0] * INV_2PI * rs;           // rho/(2*pi*sigma)

    // ---- per-row coefficients, loaded once and folded with norm
    const int mBase = rowTile * 16 + laneHi * 8;         // D: VGPR r -> row r (+8 hi half)
    float coefMu[8], coefLam[8];
    #pragma unroll
    for (int r = 0; r < 8; ++r) {
        const int   m = mBase + r;
        const float e = eps[m];
        coefMu[r]  = eps_perm[m] * e * norm;
        coefLam[r] = e * d_drop[m];
    }

    const int tCol0 = colGroup * CTW;
    for (int j = 0; j < CTW; ++j) {
        const int tCol = tCol0 + j;
        if (tCol >= colTiles) break;                     // wave-uniform

        // ---- B fragment (4x16 f32): only per-tile loads are 16 SB coords + chi
        const int   nB = tCol * 16 + laneLo;
        const float bx = sb[2 * nB + 0];
        const float by = sb[2 * nB + 1];
        v2f b;
        b.x = laneHi ? 1.0f                : bx;         // K=2 : K=0
        b.y = laneHi ? (bx * bx + by * by) : by;         // K=3 : K=1

        v8f c = {};
        c = __builtin_amdgcn_wmma_f32_16x16x4_f32(
                /*neg_a=*/false, a, /*neg_b=*/false, b,
                /*c_mod=*/(short)0, c, /*reuse_a=*/false, /*reuse_b=*/false);

        const float chin  = chi[nB];
        float*      obase = out + (size_t)mBase * (size_t)M + (size_t)nB;
        #pragma unroll
        for (int r = 0; r < 8; ++r) {
            // native v_exp_f32 (hardware exp2); log2e pre-folded into `scale`
            const float val = fmaf(coefMu[r],
                                   __builtin_amdgcn_exp2f(c[r] * scale),
                                   coefLam[r] * chin);
            __builtin_nontemporal_store(val, obase + (size_t)r * (size_t)M);
        }
    }
}

extern "C" void kernel_launch(void* const* d_in, const int* in_sizes, int n_in,
                              void* d_out, int out_size, void* d_ws, size_t ws_size,
                              hipStream_t stream) {
    const float* nuc      = (const float*)d_in[0];  // [N,2]
    const float* sb       = (const float*)d_in[1];  // [M,2]
    const float* eps_perm = (const float*)d_in[2];  // [N]
    const float* eps      = (const float*)d_in[3];  // [N]
    const float* d_drop   = (const float*)d_in[4];  // [N]
    const float* rho      = (const float*)d_in[5];  // [1]
    const float* sigma    = (const float*)d_in[6];  // [1]
    const float* chi      = (const float*)d_in[7];  // [M]
    float*       out      = (float*)d_out;          // [N,M]

    const int N = in_sizes[0] / 2;                  // 8192
    const int M = in_sizes[1] / 2;                  // 4096
    const int rowTiles = N / 16;                    // 512
    const int colTiles = M / 16;                    // 256
    const int CTW = 8;
    const int colGroups  = (colTiles + CTW - 1) / CTW;   // 32
    const int totalWaves = rowTiles * colGroups;         // 16384
    const int wavesPerBlock = 8;                         // 256 threads, wave32
    const int blocks = (totalWaves + wavesPerBlock - 1) / wavesPerBlock;

    if (M == 4096) {
        poisson_rate_kernel<4096><<<blocks, 32 * wavesPerBlock, 0, stream>>>(
            nuc, sb, eps_perm, eps, d_drop, rho, sigma, chi, out,
            M, colTiles, colGroups, totalWaves);
    } else {
        poisson_rate_kernel<0><<<blocks, 32 * wavesPerBlock, 0, stream>>>(
            nuc, sb, eps_perm, eps, d_drop, rho, sigma, chi, out,
            M, colTiles, colGroups, totalWaves);
    }
}